// HGNNLayer_5617817223406
// MI455X (gfx1250) — compile-verified
//
#include <hip/hip_runtime.h>

typedef __attribute__((ext_vector_type(2))) float v2f;
typedef __attribute__((ext_vector_type(8))) float v8f;

#define N_NODES     100000
#define N_EDGES     20000
#define NNZ_TOTAL   600000
#define DIM         128

// ---------------------------------------------------------------------------
// 1) zero scratch (deg_n, deg_e, e_feat are contiguous at ws start)
// ---------------------------------------------------------------------------
__global__ void zero_ws_kernel(float* __restrict__ p, int n) {
    int i = blockIdx.x * blockDim.x + threadIdx.x;
    if (i < n) p[i] = 0.0f;
}

// ---------------------------------------------------------------------------
// 2) degree accumulation: deg_n[node]++ , deg_e[edge]++  (float atomics)
// ---------------------------------------------------------------------------
__global__ void degree_kernel(const int* __restrict__ node_idx,
                              const int* __restrict__ edge_idx,
                              float* __restrict__ deg_n,
                              float* __restrict__ deg_e,
                              int nnz) {
    int i = blockIdx.x * blockDim.x + threadIdx.x;
    if (i >= nnz) return;
    atomicAdd(&deg_n[node_idx[i]], 1.0f);
    atomicAdd(&deg_e[edge_idx[i]], 1.0f);
}

// ---------------------------------------------------------------------------
// 3) fused dual GEMM via V_WMMA_F32_16X16X4_F32 (wave32, 16x16 tile / wave)
//    y   = x @ conv_W                      -> workspace
//    out = x @ res_W + res_b + conv_b     -> d_out (scatter adds onto this)
//
//    Block = 256 threads = 8 waves; block b owns rows [16b,16b+16),
//    wave w owns cols [16w,16w+16).  K=128 consumed in 32 steps of 4.
//
//    Fragment layouts per CDNA5 ISA (32-bit operands):
//      A (16x4):  lanes 0-15 -> M=lane, K={0,1} in v[0],v[1];
//                 lanes 16-31 -> M=lane-16, K={2,3}
//      B (4x16):  lanes 0-15 -> N=lane, K={0,1} in v[0],v[1];
//                 lanes 16-31 -> N=lane-16, K={2,3}
//      C/D (16x16): v[j] -> M = j + 8*(lane/16), N = lane%16
// ---------------------------------------------------------------------------
__global__ __launch_bounds__(256) void dual_gemm_kernel(
        const float* __restrict__ x,
        const float* __restrict__ Wc,   // conv_W  [D,D] row-major
        const float* __restrict__ Wr,   // res_W   [D,D] row-major
        const float* __restrict__ bc,   // conv_b
        const float* __restrict__ br,   // res_b
        float* __restrict__ y,          // [N,D]
        float* __restrict__ out)        // [N,D]
{
    const int wave   = threadIdx.x >> 5;       // 0..7 -> N tile
    const int lane   = threadIdx.x & 31;
    const int m0     = blockIdx.x * 16;
    const int n0     = wave * 16;
    const int sub    = lane & 15;
    const int hi     = lane >> 4;              // 0 or 1
    const int arow   = m0 + sub;               // A fragment row
    const int bcol   = n0 + sub;               // B fragment col (== output col)
    const int khalf  = hi * 2;                 // K offset {0,2} for this half-wave

    v8f acc_c = {};   // x @ conv_W
    v8f acc_r = {};   // x @ res_W

    #pragma unroll 4
    for (int k = 0; k < DIM; k += 4) {
        const int kk = k + khalf;
        // A fragment: contiguous pair x[arow][kk], x[arow][kk+1]
        v2f a = *(const v2f*)(x + arow * DIM + kk);
        // B fragments: W[kk][bcol], W[kk+1][bcol] (stride DIM)
        v2f b_c, b_r;
        b_c[0] = Wc[(kk + 0) * DIM + bcol];
        b_c[1] = Wc[(kk + 1) * DIM + bcol];
        b_r[0] = Wr[(kk + 0) * DIM + bcol];
        b_r[1] = Wr[(kk + 1) * DIM + bcol];

        acc_c = __builtin_amdgcn_wmma_f32_16x16x4_f32(
                    false, a, false, b_c, (short)0, acc_c, false, false);
        acc_r = __builtin_amdgcn_wmma_f32_16x16x4_f32(
                    false, a, false, b_r, (short)0, acc_r, false, false);
    }

    const float bias = bc[bcol] + br[bcol];    // conv_b folded into residual base
    #pragma unroll
    for (int j = 0; j < 8; ++j) {
        const int r = m0 + j + (hi << 3);
        y[r * DIM + bcol]   = acc_c[j];
        out[r * DIM + bcol] = acc_r[j] + bias;
    }
}

// ---------------------------------------------------------------------------
// 4) node -> hyperedge scatter-add (raw sum; inv_e folded into phase 5)
//    one wave per nnz entry; each lane moves 4 contiguous floats (128/32)
// ---------------------------------------------------------------------------
__global__ __launch_bounds__(256) void scatter_edge_kernel(
        const int* __restrict__ node_idx,
        const int* __restrict__ edge_idx,
        const float* __restrict__ y,
        float* __restrict__ ef,          // [E,D]
        int nnz)
{
    const int i = blockIdx.x * (blockDim.x >> 5) + (threadIdx.x >> 5);
    if (i >= nnz) return;
    const int lane = threadIdx.x & 31;
    const int n = node_idx[i];
    const int e = edge_idx[i];
    const float4 v = *(const float4*)(y + (size_t)n * DIM + lane * 4);
    float* dst = ef + (size_t)e * DIM + lane * 4;
    atomicAdd(dst + 0, v.x);
    atomicAdd(dst + 1, v.y);
    atomicAdd(dst + 2, v.z);
    atomicAdd(dst + 3, v.w);
}

// ---------------------------------------------------------------------------
// 5) hyperedge -> node scatter-add with inv_e * inv_n folded per contribution
//    (valid: both factors are constant over their segment)
// ---------------------------------------------------------------------------
__global__ __launch_bounds__(256) void scatter_node_kernel(
        const int* __restrict__ node_idx,
        const int* __restrict__ edge_idx,
        const float* __restrict__ ef,
        const float* __restrict__ deg_n,
        const float* __restrict__ deg_e,
        float* __restrict__ out,
        int nnz)
{
    const int i = blockIdx.x * (blockDim.x >> 5) + (threadIdx.x >> 5);
    if (i >= nnz) return;
    const int lane = threadIdx.x & 31;
    const int n = node_idx[i];
    const int e = edge_idx[i];
    const float dn = deg_n[n];
    const float de = deg_e[e];
    const float s = (dn > 0.0f ? 1.0f / dn : 0.0f) * (de > 0.0f ? 1.0f / de : 0.0f);
    const float4 v = *(const float4*)(ef + (size_t)e * DIM + lane * 4);
    float* dst = out + (size_t)n * DIM + lane * 4;
    atomicAdd(dst + 0, s * v.x);
    atomicAdd(dst + 1, s * v.y);
    atomicAdd(dst + 2, s * v.z);
    atomicAdd(dst + 3, s * v.w);
}

// ---------------------------------------------------------------------------
// launch
// ---------------------------------------------------------------------------
extern "C" void kernel_launch(void* const* d_in, const int* in_sizes, int n_in,
                              void* d_out, int out_size, void* d_ws, size_t ws_size,
                              hipStream_t stream) {
    const float* x        = (const float*)d_in[0];
    const int*   node_idx = (const int*)  d_in[1];
    const int*   edge_idx = (const int*)  d_in[2];
    const float* conv_W   = (const float*)d_in[3];
    const float* conv_b   = (const float*)d_in[4];
    const float* res_W    = (const float*)d_in[5];
    const float* res_b    = (const float*)d_in[6];
    float*       out      = (float*)d_out;

    // workspace layout: deg_n | deg_e | e_feat | y   (all float32)
    float* ws    = (float*)d_ws;
    float* deg_n = ws;
    float* deg_e = deg_n + N_NODES;
    float* ef    = deg_e + N_EDGES;
    float* y     = ef + (size_t)N_EDGES * DIM;

    // 1) zero deg_n + deg_e + e_feat (contiguous prefix of ws)
    const int zero_n = N_NODES + N_EDGES + N_EDGES * DIM;
    zero_ws_kernel<<<(zero_n + 255) / 256, 256, 0, stream>>>(ws, zero_n);

    // 2) degrees
    degree_kernel<<<(NNZ_TOTAL + 255) / 256, 256, 0, stream>>>(
        node_idx, edge_idx, deg_n, deg_e, NNZ_TOTAL);

    // 3) dual GEMM (WMMA): y = x@conv_W ; out = x@res_W + res_b + conv_b
    dual_gemm_kernel<<<N_NODES / 16, 256, 0, stream>>>(
        x, conv_W, res_W, conv_b, res_b, y, out);

    // 4) node -> edge aggregation (8 nnz per 256-thread block)
    scatter_edge_kernel<<<(NNZ_TOTAL + 7) / 8, 256, 0, stream>>>(
        node_idx, edge_idx, y, ef, NNZ_TOTAL);

    // 5) edge -> node aggregation with B^-1 and D^-1 normalization folded in
    scatter_node_kernel<<<(NNZ_TOTAL + 7) / 8, 256, 0, stream>>>(
        node_idx, edge_idx, ef, deg_n, deg_e, out, NNZ_TOTAL);
}